// OuterProductMean_76957224009943
// MI455X (gfx1250) — compile-verified
//
#include <hip/hip_runtime.h>
#include <hip/hip_bf16.h>

typedef __attribute__((ext_vector_type(2))) float v2f;
typedef __attribute__((ext_vector_type(8))) float v8f;

#define IN_CH 256
#define L_DIM 256
#define HID 32
#define OUT_CH 32
#define EPSV 1e-5f

// -------- workspace layout (floats) --------
// a       : [256][32]       off 0      (8192)
// s       : [256]           off 8192
// q       : [256]           off 8448
// colsum  : [32]            off 8704
// off     : [32]            off 8736
// G       : [1024][32]      off 8768   (32768)  (== [32][1024] view for GEMM B)
// P       : [256][1024]     off 41536  (262144)
#define WS_A      0
#define WS_S      8192
#define WS_Q      8448
#define WS_CS     8704
#define WS_OFF    8736
#define WS_G      8768
#define WS_P      41536

// a = x @ W1 + b1 ; s_i = sum(a_i) ; q_i = sum(a_i^2).  One block, 256 threads.
__global__ void k_a(const float* __restrict__ x, const float* __restrict__ W1,
                    const float* __restrict__ b1, float* __restrict__ a,
                    float* __restrict__ s, float* __restrict__ q) {
  __shared__ float w1s[IN_CH * HID];
  int t = threadIdx.x;
  for (int r = 0; r < 32; ++r) w1s[t + 256 * r] = W1[t + 256 * r];
  __syncthreads();
  float acc[HID];
  for (int h = 0; h < HID; ++h) acc[h] = b1[h];
  const float* xr = x + (size_t)t * IN_CH;
  for (int c = 0; c < IN_CH; ++c) {
    float xc = xr[c];
    const float* wr = w1s + c * HID;
    for (int h = 0; h < HID; ++h) acc[h] = fmaf(xc, wr[h], acc[h]);
  }
  float ss = 0.f, qq = 0.f;
  for (int h = 0; h < HID; ++h) {
    float v = acc[h];
    a[t * HID + h] = v;
    ss += v;
    qq += v * v;
  }
  s[t] = ss;
  q[t] = qq;
}

// G[k,o] = gamma[k]*W2[k,o]; colsum[o] = sum_k G; off[o] = beta.W2[:,o] + b2[o]
__global__ void k_g(const float* __restrict__ gamma, const float* __restrict__ beta,
                    const float* __restrict__ W2, const float* __restrict__ b2,
                    float* __restrict__ G, float* __restrict__ colsum,
                    float* __restrict__ off) {
  int t = threadIdx.x;
  for (int r = 0; r < 128; ++r) {
    int idx = t + 256 * r;
    G[idx] = gamma[idx >> 5] * W2[idx];
  }
  if (t < OUT_CH) {
    float cs = 0.f, of = 0.f;
    for (int k = 0; k < HID * HID; ++k) {
      float w = W2[k * OUT_CH + t];
      cs = fmaf(gamma[k], w, cs);
      of = fmaf(beta[k], w, of);
    }
    colsum[t] = cs;
    off[t] = of + b2[t];
  }
}

// P[256 x 1024] = A[256 x 32] @ G(viewed as [32 x 1024]) using f32 WMMA 16x16x4.
// grid (8 nGroups, 16 mTiles), 256 threads (8 waves); wave w -> N-tile nGroup*8+w.
__global__ void k_p(const float* __restrict__ A, const float* __restrict__ G,
                    float* __restrict__ P) {
  int lane = threadIdx.x & 31;
  int wave = threadIdx.x >> 5;
  int half = lane >> 4;
  int l15 = lane & 15;
  int m0 = blockIdx.y * 16;
  int n0 = (blockIdx.x * 8 + wave) * 16;
  int arow = m0 + l15;   // A-matrix M index (lanes 0-15 and 16-31 both map M=l15)
  int ncol = n0 + l15;   // B/D N index
  v8f acc = {};
  for (int kk = 0; kk < 8; ++kk) {
    int ka = kk * 4 + half * 2;
    v2f aF = *(const v2f*)(A + arow * HID + ka);  // K = ka, ka+1
    v2f bF;
    bF.x = G[(ka + 0) * 1024 + ncol];
    bF.y = G[(ka + 1) * 1024 + ncol];
    acc = __builtin_amdgcn_wmma_f32_16x16x4_f32(false, aF, false, bF,
                                                (short)0, acc, false, false);
  }
  for (int r = 0; r < 8; ++r)
    P[(size_t)(m0 + r + 8 * half) * 1024 + ncol] = acc[r];
}

// Per i: bil[j,o] = A[j,:] @ P_i[32x32], fused LayerNorm-collapse epilogue, store out.
// grid 256 (one block per i), 256 threads (8 waves); each wave: 2 j-tiles x 2 o-halves.
__global__ void k_out(const float* __restrict__ A, const float* __restrict__ P,
                      const float* __restrict__ s, const float* __restrict__ q,
                      const float* __restrict__ colsum, const float* __restrict__ off,
                      float* __restrict__ out) {
  int i = blockIdx.x;
  const float* Pi = P + (size_t)i * 1024;  // [e][o] row-major 32x32
  int lane = threadIdx.x & 31;
  int wave = threadIdx.x >> 5;
  int half = lane >> 4;
  int l15 = lane & 15;
  float s_i = s[i];
  float q_i = q[i];
  float cs0 = colsum[l15], cs1 = colsum[16 + l15];
  float of0 = off[l15], of1 = off[16 + l15];
  for (int rep = 0; rep < 2; ++rep) {
    int j0 = (wave + 8 * rep) * 16;
    int arow = j0 + l15;
    v8f acc0 = {}, acc1 = {};
    for (int kk = 0; kk < 8; ++kk) {
      int ka = kk * 4 + half * 2;
      v2f aF = *(const v2f*)(A + arow * HID + ka);
      v2f b0, b1;
      b0.x = Pi[(ka + 0) * OUT_CH + l15];
      b0.y = Pi[(ka + 1) * OUT_CH + l15];
      b1.x = Pi[(ka + 0) * OUT_CH + 16 + l15];
      b1.y = Pi[(ka + 1) * OUT_CH + 16 + l15];
      acc0 = __builtin_amdgcn_wmma_f32_16x16x4_f32(false, aF, false, b0,
                                                   (short)0, acc0, false, false);
      acc1 = __builtin_amdgcn_wmma_f32_16x16x4_f32(false, aF, false, b1,
                                                   (short)0, acc1, false, false);
    }
    for (int r = 0; r < 8; ++r) {
      int j = j0 + r + 8 * half;
      float mu = s_i * s[j] * (1.0f / 1024.0f);
      float var = fmaf(q_i * q[j], 1.0f / 1024.0f, -mu * mu);
      float rstd = rsqrtf(var + EPSV);
      size_t base = ((size_t)i * L_DIM + j) * OUT_CH;
      out[base + l15]      = fmaf(rstd, acc0[r] - mu * cs0, of0);
      out[base + 16 + l15] = fmaf(rstd, acc1[r] - mu * cs1, of1);
    }
  }
}

extern "C" void kernel_launch(void* const* d_in, const int* in_sizes, int n_in,
                              void* d_out, int out_size, void* d_ws, size_t ws_size,
                              hipStream_t stream) {
  (void)in_sizes; (void)n_in; (void)out_size; (void)ws_size;
  const float* x     = (const float*)d_in[0];
  const float* W1    = (const float*)d_in[1];
  const float* b1    = (const float*)d_in[2];
  const float* gamma = (const float*)d_in[3];
  const float* beta  = (const float*)d_in[4];
  const float* W2    = (const float*)d_in[5];
  const float* b2    = (const float*)d_in[6];
  float* out = (float*)d_out;
  float* w = (float*)d_ws;
  float* a  = w + WS_A;
  float* s  = w + WS_S;
  float* q  = w + WS_Q;
  float* cs = w + WS_CS;
  float* of = w + WS_OFF;
  float* G  = w + WS_G;
  float* P  = w + WS_P;

  k_a<<<1, 256, 0, stream>>>(x, W1, b1, a, s, q);
  k_g<<<1, 256, 0, stream>>>(gamma, beta, W2, b2, G, cs, of);
  k_p<<<dim3(8, 16), 256, 0, stream>>>(a, G, P);
  k_out<<<256, 256, 0, stream>>>(a, P, s, q, cs, of, out);
}